// GcnNet_6554120094284
// MI455X (gfx1250) — compile-verified
//
#include <hip/hip_runtime.h>

typedef __attribute__((ext_vector_type(2))) float v2f;
typedef __attribute__((ext_vector_type(8))) float v8f;

#define NN 2048      // N_NODES
#define DD 1024      // D
#define BB 16384     // B
#define EE 65536     // E

// ---------------------------------------------------------------- zero ----
__global__ void k_zero(float4* __restrict__ p, int n4) {
    int i = blockIdx.x * blockDim.x + threadIdx.x;
    if (i < n4) p[i] = make_float4(0.f, 0.f, 0.f, 0.f);
}

// ------------------------------------------------------------- scatter ----
// node_feat[data_x[7b+slot]] += feat[b, slot, :]   for slot in {0,1}
__global__ __launch_bounds__(256) void k_scatter(const float* __restrict__ feat,
                                                 const int* __restrict__ data_x,
                                                 float* __restrict__ node_feat) {
    int blk  = blockIdx.x;        // 0 .. 2B-1
    int b    = blk >> 1;
    int slot = blk & 1;
    int node = data_x[7 * b + slot];
    const float* src = feat + ((size_t)b * 7 + slot) * DD;
    float* dst       = node_feat + (size_t)node * DD;
    int d = threadIdx.x * 4;
    #pragma unroll
    for (int j = 0; j < 4; j++) atomicAdd(&dst[d + j], src[d + j]);
}

// ---------------------------------------------------- f32 WMMA GEMM -------
// C[M,N] = A[M,K] @ B[K,N], all row-major, M,N multiples of 128, K of 32.
// Block tile 128x128, 8 waves (waveM 0..3 x 32 rows, waveN 0..1 x 64 cols),
// each wave owns 2x4 tiles of 16x16 via v_wmma_f32_16x16x4_f32.
// Double-buffered LDS; B stored K-pair-interleaved (v2f) so B fragments are
// single conflict-free ds_load_b64s. launch_bounds(256,1) -> no spills.
#define TM 128
#define TN 128
#define TK 32
#define APAD 36     // floats/row: 144B, 16B-aligned, conflict-free strided reads
#define BP  144     // v2f/row: 128 + 16 pad -> +32 banks/row, halves disjoint

__global__ __launch_bounds__(256, 1) void k_gemm_f32(const float* __restrict__ A,
                                                     const float* __restrict__ Bm,
                                                     float* __restrict__ C,
                                                     int M, int N, int K) {
    __shared__ float Als[2][TM][APAD];       // 2 x 18 KB
    __shared__ v2f   Bls[2][TK / 2][BP];     // 2 x 18 KB

    int tid   = threadIdx.x;
    int lane  = tid & 31;
    int wid   = tid >> 5;
    int waveM = wid & 3;     // 32-row strip
    int waveN = wid >> 2;    // 64-col strip
    int bm = blockIdx.y * TM;
    int bn = blockIdx.x * TN;

    v8f acc[2][4];
    #pragma unroll
    for (int mt = 0; mt < 2; mt++)
        #pragma unroll
        for (int nt = 0; nt < 4; nt++) acc[mt][nt] = {};

    // global->reg staging addressing (strength-reduced base pointers)
    int arow = tid >> 3;          // 0..31
    int acol = (tid & 7) * 4;     // 0..28
    int bkp  = tid >> 5;          // 0..7 (pass adds 8) : K-pair index
    int bn4  = (tid & 31) * 4;    // 0..124
    const float* aptr = A + (size_t)(bm + arow) * K + acol;      // advance by k0g
    const float* bptr = Bm + (size_t)(2 * bkp) * N + bn + bn4;   // advance by k0g*N
    const size_t aRowStride = (size_t)32 * K;    // p * 32 rows
    const size_t bPassStride = (size_t)16 * N;   // p * 8 k-pairs = 16 rows

    // compute-side addressing
    int mrow = lane & 15;
    int khh  = lane >> 4;         // 0 or 1
    int khalf = khh * 2;

    float4 ra[4];
    float4 rb[2][2];

    auto gload = [&](int k0g) {
        const float* ap = aptr + k0g;
        #pragma unroll
        for (int p = 0; p < 4; p++)
            ra[p] = *(const float4*)(ap + p * aRowStride);
        const float* bp = bptr + (size_t)k0g * N;
        #pragma unroll
        for (int p = 0; p < 2; p++) {
            rb[p][0] = *(const float4*)(bp + p * bPassStride);
            rb[p][1] = *(const float4*)(bp + p * bPassStride + N);
        }
    };
    auto lstore = [&](int buf) {
        #pragma unroll
        for (int p = 0; p < 4; p++)
            *(float4*)&Als[buf][arow + p * 32][acol] = ra[p];
        #pragma unroll
        for (int p = 0; p < 2; p++) {
            v2f* dst = &Bls[buf][bkp + p * 8][bn4];
            v2f t0; t0.x = rb[p][0].x; t0.y = rb[p][1].x;
            v2f t1; t1.x = rb[p][0].y; t1.y = rb[p][1].y;
            v2f t2; t2.x = rb[p][0].z; t2.y = rb[p][1].z;
            v2f t3; t3.x = rb[p][0].w; t3.y = rb[p][1].w;
            dst[0] = t0; dst[1] = t1; dst[2] = t2; dst[3] = t3;
        }
    };
    auto compute = [&](int buf) {
        #pragma unroll
        for (int k0 = 0; k0 < TK; k0 += 4) {
            v2f af[2];
            #pragma unroll
            for (int mt = 0; mt < 2; mt++)
                af[mt] = *(const v2f*)&Als[buf][waveM * 32 + mt * 16 + mrow][k0 + khalf];
            v2f bf[4];
            int kp = (k0 >> 1) + khh;
            #pragma unroll
            for (int nt = 0; nt < 4; nt++)
                bf[nt] = Bls[buf][kp][waveN * 64 + nt * 16 + mrow];
            #pragma unroll
            for (int mt = 0; mt < 2; mt++)
                #pragma unroll
                for (int nt = 0; nt < 4; nt++)
                    acc[mt][nt] = __builtin_amdgcn_wmma_f32_16x16x4_f32(
                        false, af[mt], false, bf[nt],
                        (short)0, acc[mt][nt], false, false);
        }
    };

    // prologue: tile 0 -> buf 0
    gload(0);
    lstore(0);
    __syncthreads();

    int NK = K / TK;
    for (int kb = 0; kb < NK; kb++) {
        if (kb + 1 < NK) gload((kb + 1) * TK);   // overlap with compute
        compute(kb & 1);
        __syncthreads();
        if (kb + 1 < NK) {
            lstore((kb + 1) & 1);
            __syncthreads();
        }
    }

    // store C: VGPR v holds M=v (lanes 0-15) / M=v+8 (lanes 16-31), N=lane%16
    int rhalf = khh * 8;
    #pragma unroll
    for (int mt = 0; mt < 2; mt++)
        #pragma unroll
        for (int nt = 0; nt < 4; nt++) {
            int rbase = bm + waveM * 32 + mt * 16 + rhalf;
            int col   = bn + waveN * 64 + nt * 16 + mrow;
            #pragma unroll
            for (int v = 0; v < 8; v++)
                C[(size_t)(rbase + v) * N + col] = acc[mt][nt][v];
        }
}

// ------------------------------------------------------------- SpMM -------
// y[r,:] = sum_{e: rows[e]==r} vals[e] * x[cols[e],:]  (+bias) (+relu)
// rows sorted ascending -> binary search per row, no atomics.
__device__ __forceinline__ int lbound(const int* __restrict__ a, int n, int key) {
    int lo = 0, hi = n;
    while (lo < hi) { int mid = (lo + hi) >> 1; if (a[mid] < key) lo = mid + 1; else hi = mid; }
    return lo;
}

__global__ __launch_bounds__(256) void k_spmm_bias(const int* __restrict__ rows,
                                                   const int* __restrict__ cols,
                                                   const float* __restrict__ vals,
                                                   const float* __restrict__ x,
                                                   const float* __restrict__ bias,
                                                   float* __restrict__ y,
                                                   int width, int do_relu) {
    int r  = blockIdx.y;
    int c4 = blockIdx.x * blockDim.x + threadIdx.x;   // float4 column index
    int w4 = width >> 2;
    int lo = lbound(rows, EE, r);
    int hi = lbound(rows, EE, r + 1);

    float4 acc = make_float4(0.f, 0.f, 0.f, 0.f);
    const float4* x4 = (const float4*)x;
    for (int e = lo; e < hi; e++) {
        int   c = cols[e];
        float v = vals[e];
        float4 xv = x4[(size_t)c * w4 + c4];
        acc.x += v * xv.x; acc.y += v * xv.y; acc.z += v * xv.z; acc.w += v * xv.w;
    }
    float4 bv = ((const float4*)bias)[c4];
    acc.x += bv.x; acc.y += bv.y; acc.z += bv.z; acc.w += bv.w;
    if (do_relu) {
        acc.x = fmaxf(acc.x, 0.f); acc.y = fmaxf(acc.y, 0.f);
        acc.z = fmaxf(acc.z, 0.f); acc.w = fmaxf(acc.w, 0.f);
    }
    ((float4*)y)[(size_t)r * w4 + c4] = acc;
}

// --------------------------------------------------------- output gather --
// out[b,0]=logits[e2[b]], out[b,1]=logits[e1[b]], out[b,2:7]=feat[b,2:7]
__global__ __launch_bounds__(256) void k_out(const float* __restrict__ logits,
                                             const float* __restrict__ feat,
                                             const int* __restrict__ data_x,
                                             float* __restrict__ out) {
    int j = blockIdx.x;          // 0 .. 7B-1
    int b = j / 7;
    int s = j % 7;
    const float* src;
    if (s == 0)      src = logits + (size_t)data_x[7 * b + 1] * DD;  // e2
    else if (s == 1) src = logits + (size_t)data_x[7 * b + 0] * DD;  // e1
    else             src = feat + ((size_t)b * 7 + s) * DD;
    float* dst = out + ((size_t)b * 7 + s) * DD;
    ((float4*)dst)[threadIdx.x] = ((const float4*)src)[threadIdx.x];
}

// ------------------------------------------------------------ launcher ----
extern "C" void kernel_launch(void* const* d_in, const int* in_sizes, int n_in,
                              void* d_out, int out_size, void* d_ws, size_t ws_size,
                              hipStream_t stream) {
    const float* feat     = (const float*)d_in[0];
    const int*   data_x   = (const int*)d_in[1];
    const int*   adj_rows = (const int*)d_in[2];
    const int*   adj_cols = (const int*)d_in[3];
    const float* adj_vals = (const float*)d_in[4];
    const float* W1 = (const float*)d_in[5];
    const float* b1 = (const float*)d_in[6];
    const float* W2 = (const float*)d_in[7];
    const float* b2 = (const float*)d_in[8];
    const float* W3 = (const float*)d_in[9];
    const float* b3 = (const float*)d_in[10];
    float* out = (float*)d_out;

    float* bufA = (float*)d_ws;                       // 16 MB
    float* bufB = bufA + (size_t)NN * NN;             // 16 MB

    // 1. zero node accumulator (2048x1024)
    {
        int n4 = NN * DD / 4;
        k_zero<<<(n4 + 255) / 256, 256, 0, stream>>>((float4*)bufA, n4);
    }
    // 2. scatter feat[:,0:2,:] into node_feat (bufA)
    k_scatter<<<2 * BB, 256, 0, stream>>>(feat, data_x, bufA);

    // 3. T1 = node_feat @ W1  (2048x1024 @ 1024x2048 -> bufB 2048x2048)
    k_gemm_f32<<<dim3(NN / TN, NN / TM), 256, 0, stream>>>(bufA, W1, bufB, NN, NN, DD);
    // 4. H1 = relu(spmm(T1) + b1) -> bufA (2048x2048)
    k_spmm_bias<<<dim3(NN / 1024, NN), 256, 0, stream>>>(adj_rows, adj_cols, adj_vals,
                                                         bufB, b1, bufA, NN, 1);
    // 5. T2 = H1 @ W2 -> bufB (2048x2048)
    k_gemm_f32<<<dim3(NN / TN, NN / TM), 256, 0, stream>>>(bufA, W2, bufB, NN, NN, NN);
    // 6. H2 = spmm(T2) + b2 -> bufA (2048x2048)
    k_spmm_bias<<<dim3(NN / 1024, NN), 256, 0, stream>>>(adj_rows, adj_cols, adj_vals,
                                                         bufB, b2, bufA, NN, 0);
    // 7. T3 = H2 @ W3 -> bufB (2048x1024)
    k_gemm_f32<<<dim3(DD / TN, NN / TM), 256, 0, stream>>>(bufA, W3, bufB, NN, DD, NN);
    // 8. logits = spmm(T3) + b3 -> bufA (2048x1024)
    k_spmm_bias<<<dim3(DD / 1024, NN), 256, 0, stream>>>(adj_rows, adj_cols, adj_vals,
                                                         bufB, b3, bufA, DD, 0);
    // 9. gather output (B,7,D)
    k_out<<<7 * BB, 256, 0, stream>>>(bufA, feat, data_x, out);
}